// MultiHeadAttention_spatial_self_70866960384612
// MI455X (gfx1250) — compile-verified
//
#include <hip/hip_runtime.h>
#include <hip/hip_bf16.h>
#include <math.h>

#define NN 15
#define HEADS 3
#define TOK_PER_BLK 64
#define THREADS 64   // 2 waves (wave32)

typedef float v2f __attribute__((ext_vector_type(2)));
typedef float v8f __attribute__((ext_vector_type(8)));

__global__ __launch_bounds__(THREADS)
void mhsa_spatial_wmma_kernel(const float* __restrict__ q,
                              const float* __restrict__ k,
                              const float* __restrict__ v,
                              const float* __restrict__ wq, const float* __restrict__ bq,
                              const float* __restrict__ wk, const float* __restrict__ bk,
                              const float* __restrict__ wv, const float* __restrict__ bv,
                              const float* __restrict__ wo, const float* __restrict__ bo,
                              const float* __restrict__ graph,
                              float* __restrict__ out, int nTok)
{
    __shared__ float sQh[TOK_PER_BLK * 48];   // [tok][h*16 + n]; reused as output staging
    __shared__ float sKh[TOK_PER_BLK * 48];
    __shared__ float sVh[TOK_PER_BLK * 48];
    __shared__ float sCat[TOK_PER_BLK * 48];  // concat rows, K-padded 45->48
    __shared__ float sWo[48 * 48];            // wo zero-padded to 48x48, row-major [j][k]
    __shared__ int   sEdge[NN * NN];
    __shared__ int   sEdgeCnt;

    const int tid     = threadIdx.x;
    const int blkBase = blockIdx.x * TOK_PER_BLK;
    const int tok     = blkBase + tid;
    const int tokC    = (tok < nTok) ? tok : (nTok - 1);   // clamp; keep EXEC full

    // --- stage zero-padded wo into LDS (cooperative, coalesced) ---
    for (int idx = tid; idx < 48 * 48; idx += THREADS) {
        int j = idx / 48, kk = idx % 48;
        sWo[idx] = (j < 45 && kk < 45) ? wo[j * 45 + kk] : 0.0f;
    }
    // --- compact graph mask into edge list (deterministic serial order) ---
    if (tid == 0) {
        int c = 0;
#pragma clang loop unroll(disable)
        for (int n = 0; n < NN; ++n)
#pragma clang loop unroll(disable)
            for (int m = 0; m < NN; ++m) {
                float g = graph[n * NN + m] + graph[225 + n * NN + m] + graph[450 + n * NN + m];
                if (g != 0.0f) sEdge[c++] = (n << 4) | m;
            }
        sEdgeCnt = c;
    }
    __syncthreads();

    // ================= per-token projections =================
    {
        float x[45];
        const float* qp = q + (size_t)tokC * 45;
#pragma unroll
        for (int i = 0; i < 45; ++i) x[i] = qp[i];
#pragma unroll
        for (int n = 0; n < NN; ++n) {
            float x0 = x[n], x1 = x[15 + n], x2 = x[30 + n];
#pragma unroll
            for (int h = 0; h < HEADS; ++h)
                sQh[tid * 48 + h * 16 + n] =
                    wq[h * 3 + 0] * x0 + wq[h * 3 + 1] * x1 + wq[h * 3 + 2] * x2 + bq[h];
        }
        const float* kp = k + (size_t)tokC * 45;
#pragma unroll
        for (int i = 0; i < 45; ++i) x[i] = kp[i];
#pragma unroll
        for (int n = 0; n < NN; ++n) {
            float x0 = x[n], x1 = x[15 + n], x2 = x[30 + n];
#pragma unroll
            for (int h = 0; h < HEADS; ++h)
                sKh[tid * 48 + h * 16 + n] =
                    wk[h * 3 + 0] * x0 + wk[h * 3 + 1] * x1 + wk[h * 3 + 2] * x2 + bk[h];
        }
        const float* vp = v + (size_t)tokC * 45;
#pragma unroll
        for (int i = 0; i < 45; ++i) x[i] = vp[i];
#pragma unroll
        for (int n = 0; n < NN; ++n) {
            float x0 = x[n], x1 = x[15 + n], x2 = x[30 + n];
#pragma unroll
            for (int h = 0; h < HEADS; ++h)
                sVh[tid * 48 + h * 16 + n] =
                    wv[h * 3 + 0] * x0 + wv[h * 3 + 1] * x1 + wv[h * 3 + 2] * x2 + bv[h];
        }
    }

    // zero own concat row (incl. K padding 45..47)
#pragma unroll
    for (int i = 0; i < 48; ++i) sCat[tid * 48 + i] = 0.0f;

    // ================= masked flattened softmax + AV =================
    const int ec = sEdgeCnt;
    for (int h = 0; h < HEADS; ++h) {
        const float* qh = &sQh[tid * 48 + h * 16];
        const float* kh = &sKh[tid * 48 + h * 16];
        const float* vh = &sVh[tid * 48 + h * 16];
        // max over unmasked scores; all scores = |.| >= 0 so init 0 is exact
        float mx = 0.0f;
        for (int e = 0; e < ec; ++e) {
            int nm = sEdge[e];
            float s = fabsf(qh[nm >> 4] * kh[nm & 15]);
            mx = fmaxf(mx, s);
        }
        // shared denominator over the flattened 15x15 grid
        float S = 0.0f;
        for (int e = 0; e < ec; ++e) {
            int nm = sEdge[e];
            int n = nm >> 4, m = nm & 15;
            float w = __expf(fabsf(qh[n] * kh[m]) - mx);
            S += w;
            sCat[tid * 48 + n * 3 + h] += w * vh[m];
        }
        float inv = 1.0f / S;
#pragma unroll
        for (int n = 0; n < NN; ++n) sCat[tid * 48 + n * 3 + h] *= inv;
    }
    __syncthreads();

    // ================= output projection via V_WMMA_F32_16X16X4_F32 =================
    // y(64x45) = sCat(64x48) * wo^T + bo ; K padded to 48 (12 steps of 4)
    // D tiles are staged back into LDS (sQh is dead now) so the global write
    // below is a single fully coalesced contiguous stream — no divergence here.
    const int wave  = tid >> 5;
    const int lane  = tid & 31;
    const int lrow  = lane & 15;   // M (for A/D) or N (for B/D)
    const int khalf = lane >> 4;   // selects K pair {0,1} vs {2,3}
    float* sOut = sQh;             // reuse: [tok][j] stride 48, cols 45..47 scratch

    for (int mt = 0; mt < 2; ++mt) {
        const int mbase = wave * 32 + mt * 16;
        // hoist A fragments for all 12 K-steps (ISA 16x4 f32 A layout)
        v2f aF[12];
#pragma unroll
        for (int ks = 0; ks < 12; ++ks) {
            int k0 = ks * 4 + 2 * khalf;
            aF[ks][0] = sCat[(mbase + lrow) * 48 + k0];
            aF[ks][1] = sCat[(mbase + lrow) * 48 + k0 + 1];
        }
        for (int nt = 0; nt < 3; ++nt) {
            const int nbase = nt * 16;
            const int jn = nbase + lrow;               // output column (45..47 = pad)
            float bias = (jn < 45) ? bo[jn] : 0.0f;    // C depends only on N
            v8f c;
#pragma unroll
            for (int r = 0; r < 8; ++r) c[r] = bias;
#pragma unroll
            for (int ks = 0; ks < 12; ++ks) {
                int k0 = ks * 4 + 2 * khalf;
                v2f bF;
                bF[0] = sWo[jn * 48 + k0];             // B[k][n] = wo[jn][k]
                bF[1] = sWo[jn * 48 + k0 + 1];
                c = __builtin_amdgcn_wmma_f32_16x16x4_f32(
                        false, aF[ks], false, bF, (short)0, c, false, false);
            }
            // unguarded D -> LDS (pad columns land in sOut scratch)
#pragma unroll
            for (int r = 0; r < 8; ++r) {
                int M = r + 8 * khalf;                 // ISA C/D layout
                sOut[(mbase + M) * 48 + jn] = c[r];
            }
        }
    }
    __syncthreads();

    // coalesced contiguous write of the block's 64x45 outputs
    const size_t gBase = (size_t)blkBase * 45;
    for (int idx = tid; idx < TOK_PER_BLK * 45; idx += THREADS) {
        int t = idx / 45, j = idx - t * 45;
        if (blkBase + t < nTok) out[gBase + idx] = sOut[t * 48 + j];
    }
}

extern "C" void kernel_launch(void* const* d_in, const int* in_sizes, int n_in,
                              void* d_out, int out_size, void* d_ws, size_t ws_size,
                              hipStream_t stream) {
    const float* q     = (const float*)d_in[0];
    const float* k     = (const float*)d_in[1];
    const float* v     = (const float*)d_in[2];
    const float* wq    = (const float*)d_in[3];
    const float* bq    = (const float*)d_in[4];
    const float* wk    = (const float*)d_in[5];
    const float* bk    = (const float*)d_in[6];
    const float* wv    = (const float*)d_in[7];
    const float* bv    = (const float*)d_in[8];
    const float* wo    = (const float*)d_in[9];
    const float* bo    = (const float*)d_in[10];
    const float* graph = (const float*)d_in[11];
    float* out = (float*)d_out;

    const int nTok = in_sizes[0] / 45;                     // 32*2048 = 65536
    const int blocks = (nTok + TOK_PER_BLK - 1) / TOK_PER_BLK;

    mhsa_spatial_wmma_kernel<<<dim3(blocks), dim3(THREADS), 0, stream>>>(
        q, k, v, wq, bq, wk, bk, wv, bv, wo, bo, graph, out, nTok);
}